// VertiHoriMedian_40690520162641
// MI455X (gfx1250) — compile-verified
//
#include <hip/hip_runtime.h>
#include <stdint.h>

// Problem geometry (fixed by the reference)
#define H_     2160
#define W_     3840
#define C_     3
#define CROP_  3
#define N_     (H_ - 2 * CROP_)   // 2154 output rows
#define WF_    (W_ * C_)          // 11520 flattened columns (W*C)

// Tiling
#define TH       16               // output rows per block
#define TW       64               // output flattened cols per block (WF_ % TW == 0)
#define HALO_T   4                // rows above   (needs r-4)
#define HALO_B   2                // rows below   (needs r+2)
#define HALO_L   12               // 4 pixel cols * 3 channels
#define HALO_R   6                // 2 pixel cols * 3 channels
#define LDS_H    (TH + HALO_T + HALO_B)   // 22
#define LDS_W    (TW + HALO_L + HALO_R)   // 82
#define NTHREADS 256              // 8 wave32

typedef __attribute__((ext_vector_type(4))) float v4f;

__device__ __forceinline__ void cswap(float& a, float& b) {
    float lo = fminf(a, b);
    float hi = fmaxf(a, b);
    a = lo; b = hi;
}

// Oblivious insertion-sort network on 5 elements; middle element is the median.
__device__ __forceinline__ float median5(float a0, float a1, float a2, float a3, float a4) {
    cswap(a0, a1);
    cswap(a1, a2); cswap(a0, a1);
    cswap(a2, a3); cswap(a1, a2); cswap(a0, a1);
    cswap(a3, a4); cswap(a2, a3); cswap(a1, a2); cswap(a0, a1);
    return a2;
}

__device__ __forceinline__ float median3(float a, float b, float c) {
#if __has_builtin(__builtin_amdgcn_fmed3f)
    return __builtin_amdgcn_fmed3f(a, b, c);   // v_med3_f32 (inputs are NaN-free)
#else
    return fmaxf(fminf(a, b), fminf(fmaxf(a, b), c));
#endif
}

__global__ __launch_bounds__(NTHREADS)
void VertiHoriMedian_kernel(const float* __restrict__ img, float* __restrict__ out) {
    __shared__ float sIn[LDS_H * LDS_W];   // clamped input tile (x = img[CROP:-CROP])
    __shared__ float sV2[TH * LDS_W];      // result after the full vertical pipeline

    const int tid = threadIdx.x;
    const int f0  = blockIdx.x * TW;       // flattened-column tile base
    const int r0  = blockIdx.y * TH;       // output-row tile base

    // ---- Stage A: async DMA of the clamped input tile into LDS (CDNA5 path) ----
    // LDS col lf corresponds to flattened col (f0 - 12 + lf); since the halo is
    // exactly -4 pixel columns, pixelcol = (f0+lf)/3 - 4, channel = (f0+lf)%3.
    for (int t = tid; t < LDS_H * LDS_W; t += NTHREADS) {
        int ty = t / LDS_W;
        int lf = t - ty * LDS_W;
        int gr = r0 - HALO_T + ty;
        gr = max(0, min(gr, N_ - 1));                 // row edge clamp
        int g2 = f0 + lf;                             // >= 0 always
        int cc = g2 / 3 - 4;
        int ch = g2 - (g2 / 3) * 3;
        cc = max(0, min(cc, W_ - 1));                 // per-channel column clamp
        // byte offset into img (whole image ~99MB, fits u32)
        uint32_t goff   = (uint32_t)(((uint32_t)(gr + CROP_) * (uint32_t)WF_ +
                                      (uint32_t)(cc * 3 + ch)) * 4u);
        // low 32 bits of a generic LDS pointer == DS byte address
        uint32_t ldsOff = (uint32_t)(uintptr_t)(&sIn[t]);
        // GVS mode: copies 4 bytes from global address saddr plus offset vaddr to the LDS address in vdst; tracked by ASYNCcnt
        asm volatile("global_load_async_to_lds_b32 %0, %1, %2"
                     :: "v"(ldsOff), "v"(goff), "s"(img)
                     : "memory");
    }
#if __has_builtin(__builtin_amdgcn_s_wait_asynccnt)
    __builtin_amdgcn_s_wait_asynccnt(0);
#else
    asm volatile("s_wait_asynccnt 0" ::: "memory");
#endif
    __syncthreads();

    // ---- Stage B: vertical pipeline (median5, shift+median3) -> sV2 ----
    // v2[r] = median3( (r>=2)? v[r-2] : 0,
    //                  (r>=1)? v[r-1] : 0,
    //                  v[min(r, N-2)] ),  v[k] = med5 over clamped rows k-2..k+2.
    for (int t = tid; t < TH * LDS_W; t += NTHREADS) {
        int tr = t / LDS_W;
        int lf = t - tr * LDS_W;
        int r  = r0 + tr;
        const float* col = &sIn[lf];
        float x0 = col[(tr + 0) * LDS_W];
        float x1 = col[(tr + 1) * LDS_W];
        float x2 = col[(tr + 2) * LDS_W];
        float x3 = col[(tr + 3) * LDS_W];
        float x4 = col[(tr + 4) * LDS_W];
        float x5 = col[(tr + 5) * LDS_W];
        float x6 = col[(tr + 6) * LDS_W];
        float m0 = median5(x0, x1, x2, x3, x4);   // v[r-2]
        float m1 = median5(x1, x2, x3, x4, x5);   // v[r-1]
        float m2 = median5(x2, x3, x4, x5, x6);   // v[r]
        float a = (r >= 2) ? m0 : 0.0f;
        float b = (r >= 1) ? m1 : 0.0f;
        float g = (r == N_ - 1) ? m1 : m2;
        sV2[t] = median3(a, b, g);
    }
    __syncthreads();

    // ---- Stage C: horizontal pipeline from sV2, vectorized NT stores ----
    // 4 consecutive flattened cols per thread -> one 16B-aligned b128 store.
    // TH * (TW/4) == NTHREADS: exactly one tile element group per thread.
    {
        int t  = tid;                              // t in [0, 256)
        int tr = t / (TW / 4);
        int tq = t - tr * (TW / 4);
        int r  = r0 + tr;
        if (r < N_) {
            int fbase = f0 + tq * 4;
            v4f res;
#pragma unroll
            for (int k = 0; k < 4; ++k) {
                int f  = fbase + k;
                int tf = tq * 4 + k;
                const float* row = &sV2[tr * LDS_W + tf]; // row[0]: pixel col c-4 (clamped)
                float y0 = row[0],  y1 = row[3],  y2 = row[6],  y3 = row[9];
                float y4 = row[12], y5 = row[15], y6 = row[18];
                float m0 = median5(y0, y1, y2, y3, y4);   // h1[c-2]
                float m1 = median5(y1, y2, y3, y4, y5);   // h1[c-1]
                float m2 = median5(y2, y3, y4, y5, y6);   // h1[c]
                int c = f / 3;
                float a = (c >= 2) ? m0 : 0.0f;
                float b = (c >= 1) ? m1 : 0.0f;
                float g = (c == W_ - 1) ? m1 : m2;
                res[k] = median3(a, b, g);
            }
            // Streaming store: keep L2 reserved for img halo reuse.
            __builtin_nontemporal_store(res, (v4f*)(out + (size_t)r * WF_ + (size_t)fbase));
        }
    }
}

extern "C" void kernel_launch(void* const* d_in, const int* in_sizes, int n_in,
                              void* d_out, int out_size, void* d_ws, size_t ws_size,
                              hipStream_t stream) {
    const float* img = (const float*)d_in[0];
    // d_in[1] (mask) and d_in[2] (vertical_size==5) are unused by the reference math.
    float* out = (float*)d_out;
    (void)in_sizes; (void)n_in; (void)out_size; (void)d_ws; (void)ws_size;

    dim3 grid(WF_ / TW, (N_ + TH - 1) / TH);   // 180 x 135
    VertiHoriMedian_kernel<<<grid, NTHREADS, 0, stream>>>(img, out);
}